// EdgeConvLayer_4398046511915
// MI455X (gfx1250) — compile-verified
//
#include <hip/hip_runtime.h>

typedef __attribute__((ext_vector_type(2))) float v2f;
typedef __attribute__((ext_vector_type(8))) float v8f;

#define WAVES_PER_BLOCK 4
#define BLOCK_THREADS   128
#define ROW_STRIDE      132    // 128 + 4 floats pad -> conflict-free ds_load_b64
#define BN_EPS          1e-5f
#define NEG_SLOPE       0.3f

// One wave computes h[tile 16 edges][64 ch] = [x_i | x_j - x_i] @ W^T
// into four 16x16 f32 accumulator tiles using V_WMMA_F32_16X16X4_F32.
__device__ inline void edge_tile_gemm(const float* __restrict__ feat,
                                      const long long* __restrict__ ei,
                                      long long E, long long tileBase,
                                      const float* __restrict__ W,
                                      float* __restrict__ smemWave,
                                      v8f C[4])
{
    const int lane = threadIdx.x & 31;
    const int r = lane & 15;      // edge row within tile (A/M, and B/N column)
    const int g = lane >> 4;      // lane group selects K sub-pair

    long long e = tileBase + r;
    if (e >= E) e = E - 1;        // clamp; invalid rows masked by callers
    const long long sI = ei[e];         // src -> x_j
    const long long dI = ei[E + e];     // dst -> x_i
    const float* fs = feat + sI * 64;
    const float* fd = feat + dI * 64;

    // Stage M-tile row r: cols [0,64) = x_i ; cols [64,128) = x_j - x_i
    float* row = smemWave + r * ROW_STRIDE;
    if (g == 0) {
        for (int c = 0; c < 64; c += 4) {
            *(float4*)(row + c) = *(const float4*)(fd + c);
        }
    } else {
        for (int c = 0; c < 64; c += 4) {
            float4 a = *(const float4*)(fs + c);
            float4 b = *(const float4*)(fd + c);
            float4 d;
            d.x = a.x - b.x; d.y = a.y - b.y; d.z = a.z - b.z; d.w = a.w - b.w;
            *(float4*)(row + 64 + c) = d;
        }
    }
    __syncthreads();

    // A fragments: lane holds M=r, K = 4*kk + 2*g + {0,1}  (16x4 f32 layout)
    v2f A[32];
    const float* myRow = smemWave + r * ROW_STRIDE + 2 * g;
#pragma unroll
    for (int kk = 0; kk < 32; ++kk) {
        A[kk] = *(const v2f*)(myRow + 4 * kk);
    }

#pragma unroll
    for (int nt = 0; nt < 4; ++nt) {
        v8f acc = {0.f, 0.f, 0.f, 0.f, 0.f, 0.f, 0.f, 0.f};
        // B fragment: lane holds N = nt*16 + r, K = 4*kk + 2*g + {0,1}
        const float* wrow = W + (nt * 16 + r) * 128 + 2 * g;
#pragma unroll
        for (int kk = 0; kk < 32; ++kk) {
            v2f b = *(const v2f*)(wrow + 4 * kk);
            acc = __builtin_amdgcn_wmma_f32_16x16x4_f32(
                false, A[kk], false, b, (short)0, acc, false, false);
        }
        C[nt] = acc;
    }
    // C/D layout: lane holds N = r (channel), M = v + 8*g (edge row), v = 0..7
}

__global__ __launch_bounds__(BLOCK_THREADS)
void k1_gemm_stats(const float* __restrict__ feat,
                   const long long* __restrict__ ei,
                   const float* __restrict__ W,
                   const float* __restrict__ bias,
                   long long E, int numTiles,
                   float* __restrict__ partials)
{
    __shared__ float stage[WAVES_PER_BLOCK][16 * ROW_STRIDE];
    __shared__ float blk[128];   // [0,64) sum, [64,128) sumsq

    const int tid = threadIdx.x;
    if (tid < 128) blk[tid] = 0.f;   // visible after gemm's internal barrier

    const int wave = tid >> 5;
    const int lane = tid & 31;
    const int tile = blockIdx.x * WAVES_PER_BLOCK + wave;
    const int tileC = (tile < numTiles) ? tile : (numTiles - 1);
    const long long tileBase = (long long)tileC * 16;

    v8f C[4];
    edge_tile_gemm(feat, ei, E, tileBase, W, &stage[wave][0], C);

    const int r = lane & 15, g = lane >> 4;
    if (tile < numTiles) {
#pragma unroll
        for (int nt = 0; nt < 4; ++nt) {
            const int ch = nt * 16 + r;
            const float bv = bias[ch];
            float s = 0.f, ss = 0.f;
#pragma unroll
            for (int v = 0; v < 8; ++v) {
                const long long e = tileBase + v + 8 * g;
                const float h = C[nt][v] + bv;
                const float m = (e < E) ? 1.f : 0.f;
                s  += m * h;
                ss += m * h * h;
            }
            atomicAdd(&blk[ch], s);        // ds_add_f32
            atomicAdd(&blk[64 + ch], ss);
        }
    }
    __syncthreads();
    if (tid < 128) partials[(long long)blockIdx.x * 128 + tid] = blk[tid];
}

__global__ __launch_bounds__(256)
void k2_finalize(const float* __restrict__ partials, int nblk, long long E,
                 const float* __restrict__ gamma, const float* __restrict__ beta,
                 const float* __restrict__ bias,
                 float* __restrict__ scale, float* __restrict__ shift)
{
    __shared__ float red0[256];
    __shared__ float red1[256];
    const int ch = blockIdx.x;   // 0..63
    float s = 0.f, ss = 0.f;
    for (int i = threadIdx.x; i < nblk; i += 256) {
        s  += partials[(long long)i * 128 + ch];
        ss += partials[(long long)i * 128 + 64 + ch];
    }
    red0[threadIdx.x] = s;
    red1[threadIdx.x] = ss;
    __syncthreads();
    for (int off = 128; off > 0; off >>= 1) {
        if (threadIdx.x < off) {
            red0[threadIdx.x] += red0[threadIdx.x + off];
            red1[threadIdx.x] += red1[threadIdx.x + off];
        }
        __syncthreads();
    }
    if (threadIdx.x == 0) {
        const float inv_n = 1.f / (float)E;
        const float mean = red0[0] * inv_n;
        const float var  = red1[0] * inv_n - mean * mean;
        const float sc   = gamma[ch] * rsqrtf(var + BN_EPS);
        scale[ch] = sc;
        // fold conv bias into shift: (h_raw + b - mean)*sc + beta
        shift[ch] = beta[ch] + (bias[ch] - mean) * sc;
    }
}

__global__ __launch_bounds__(BLOCK_THREADS)
void k3_apply_scatter(const float* __restrict__ feat,
                      const long long* __restrict__ ei,
                      const float* __restrict__ W,
                      const float* __restrict__ scale,
                      const float* __restrict__ shift,
                      long long E, int numTiles,
                      float* __restrict__ outAccum, float* __restrict__ cnt)
{
    __shared__ float stage[WAVES_PER_BLOCK][16 * ROW_STRIDE];
    const int wave = threadIdx.x >> 5;
    const int lane = threadIdx.x & 31;
    const int tile = blockIdx.x * WAVES_PER_BLOCK + wave;
    const int tileC = (tile < numTiles) ? tile : (numTiles - 1);
    const long long tileBase = (long long)tileC * 16;

    v8f C[4];
    edge_tile_gemm(feat, ei, E, tileBase, W, &stage[wave][0], C);
    if (tile >= numTiles) return;   // after the only barrier -> safe

    const int r = lane & 15, g = lane >> 4;
    float sc[4], sh[4];
#pragma unroll
    for (int nt = 0; nt < 4; ++nt) {
        const int ch = nt * 16 + r;
        sc[nt] = scale[ch];
        sh[nt] = shift[ch];
    }

#pragma unroll
    for (int v = 0; v < 8; ++v) {
        const long long e = tileBase + v + 8 * g;  // edge row M = v + 8g
        if (e < E) {
            const long long d = ei[E + e];         // dst node
            float* base = outAccum + d * 64;
#pragma unroll
            for (int nt = 0; nt < 4; ++nt) {
                float h = C[nt][v] * sc[nt] + sh[nt];
                h = (h >= 0.f) ? h : NEG_SLOPE * h;  // LeakyReLU(0.3)
                atomicAdd(base + nt * 16 + r, h);
            }
        }
    }
    if (g == 0) {   // one count per edge row
        const long long e = tileBase + r;
        if (e < E) atomicAdd(cnt + ei[E + e], 1.0f);
    }
}

__global__ __launch_bounds__(256)
void k4_divide(float* __restrict__ out, const float* __restrict__ cnt,
               long long total)
{
    const long long i = (long long)blockIdx.x * 256 + threadIdx.x;
    if (i < total) {
        const float c = cnt[i >> 6];
        out[i] = out[i] / fmaxf(c, 1.0f);
    }
}

extern "C" void kernel_launch(void* const* d_in, const int* in_sizes, int n_in,
                              void* d_out, int out_size, void* d_ws, size_t ws_size,
                              hipStream_t stream)
{
    const float*      feat  = (const float*)d_in[0];
    const long long*  ei    = (const long long*)d_in[1];  // int64 [2, E]
    const float*      W     = (const float*)d_in[2];      // [64, 128]
    const float*      bias  = (const float*)d_in[3];      // [64]
    const float*      gamma = (const float*)d_in[4];
    const float*      beta  = (const float*)d_in[5];
    float*            out   = (float*)d_out;              // [N, 64]

    const long long N = in_sizes[0] / 64;
    const long long E = in_sizes[1] / 2;
    const int numTiles = (int)((E + 15) / 16);
    const int nblk = (numTiles + WAVES_PER_BLOCK - 1) / WAVES_PER_BLOCK;

    float* ws       = (float*)d_ws;
    float* partials = ws;                                   // nblk*128 floats
    float* scale    = partials + (size_t)nblk * 128;        // 64
    float* shift    = scale + 64;                           // 64
    float* cnt      = shift + 64;                           // N floats

    hipMemsetAsync(d_out, 0, (size_t)out_size * sizeof(float), stream);
    hipMemsetAsync(cnt, 0, (size_t)N * sizeof(float), stream);

    k1_gemm_stats<<<nblk, BLOCK_THREADS, 0, stream>>>(
        feat, ei, W, bias, E, numTiles, partials);

    k2_finalize<<<64, 256, 0, stream>>>(
        partials, nblk, E, gamma, beta, bias, scale, shift);

    k3_apply_scatter<<<nblk, BLOCK_THREADS, 0, stream>>>(
        feat, ei, W, scale, shift, E, numTiles, out, cnt);

    const long long total = N * 64;
    k4_divide<<<(int)((total + 255) / 256), 256, 0, stream>>>(out, cnt, total);
}